// UniEncoder_72249939853829
// MI455X (gfx1250) — compile-verified
//
#include <hip/hip_runtime.h>
#include <hip/hip_bf16.h>

typedef float v2f __attribute__((ext_vector_type(2)));
typedef float v8f __attribute__((ext_vector_type(8)));
typedef int   v4i __attribute__((ext_vector_type(4)));

#define Bsz 32
#define Lsz 1024
#define Hsz 768
#define Ssz 16
#define HIST_CODE 0x40000000

// ---------------------------------------------------------------------------
// Kernel 1: per-batch prep. valid[b] = sum(attention_mask[b]); zero logit acc;
// codes[b][l] = -1 (invalid) | HIST_CODE (l < hist) | segment id.
// ---------------------------------------------------------------------------
__global__ __launch_bounds__(256) void prep_kernel(const int* __restrict__ amask,
                                                   const int* __restrict__ bos,
                                                   int* __restrict__ valid,
                                                   int* __restrict__ codes,
                                                   float* __restrict__ acc) {
    const int b = blockIdx.x;
    const int tid = threadIdx.x;
    __shared__ int sdata[256];
    __shared__ int sbos[Ssz];
    __shared__ int svalid;

    int sum = 0;
    for (int l = tid; l < Lsz; l += 256) sum += amask[b * Lsz + l];
    sdata[tid] = sum;
    __syncthreads();
    for (int s2 = 128; s2 > 0; s2 >>= 1) {
        if (tid < s2) sdata[tid] += sdata[tid + s2];
        __syncthreads();
    }
    if (tid == 0) { svalid = sdata[0]; valid[b] = sdata[0]; }
    if (tid < Ssz) {
        sbos[tid] = bos[b * Ssz + tid];
        acc[b * Ssz + tid] = 0.0f;
    }
    __syncthreads();

    const int v = svalid;
    const int hist = sbos[0];
    for (int l = tid; l < Lsz; l += 256) {
        int code;
        if (l >= v) {
            code = -1;                       // outside valid region
        } else if (l < hist) {
            code = HIST_CODE;                // history prefix
        } else {
            int seg = -1;
            #pragma unroll
            for (int s = 0; s < Ssz; ++s) seg += (l >= sbos[s]) ? 1 : 0;
            code = seg;                      // searchsorted(bos, l, right) - 1
        }
        codes[b * Lsz + l] = code;
    }
}

// ---------------------------------------------------------------------------
// Kernel 2: streaming mask writer. One block per (row i, batch b); each thread
// emits one v4i (4 columns) with a nontemporal 16B store. 128 MB total.
// ---------------------------------------------------------------------------
__device__ __forceinline__ int mask_val(int ci, int cj) {
    return (ci >= 0 && cj >= 0 &&
            (ci == HIST_CODE || cj == HIST_CODE || ci == cj)) ? 1 : 0;
}

__global__ __launch_bounds__(256) void mask_kernel(const int* __restrict__ codes,
                                                   int* __restrict__ outmask) {
    const int i = blockIdx.x;   // row in [0, L)
    const int b = blockIdx.y;   // batch
    const int ci = codes[b * Lsz + i];

    const v4i* cj4 = (const v4i*)(codes + b * Lsz);
    v4i* out = (v4i*)(outmask + ((size_t)b * Lsz + i) * Lsz);

    v4i r;
    if (ci < 0) {
        r = (v4i){0, 0, 0, 0};               // invalid row: all zeros
    } else {
        const v4i cj = cj4[threadIdx.x];
        r.x = mask_val(ci, cj.x);
        r.y = mask_val(ci, cj.y);
        r.z = mask_val(ci, cj.z);
        r.w = mask_val(ci, cj.w);
    }
    __builtin_nontemporal_store(r, out + threadIdx.x);
}

// ---------------------------------------------------------------------------
// Kernel 3: segment-sum pooling via V_WMMA_F32_16X16X4_F32.
// One wave per (batch, 16-wide H tile). D[s][n] accumulates
// sum_k ind(k, s) * toks[b][k][h0+n] over K = L = 1024 in steps of 4.
// A (16x4 f32) ISA layout: lanes 0-15 -> {K0,K1}, lanes 16-31 -> {K2,K3}.
// B (4x16 f32): one K-row per 16-lane half per VGPR.
// C/D (16x16 f32): VGPR v holds M = v + 8*(lane>=16), N = lane&15.
// ---------------------------------------------------------------------------
__global__ __launch_bounds__(32) void pool_wmma_kernel(const float* __restrict__ toks,
                                                       const int* __restrict__ bos,
                                                       const int* __restrict__ valid,
                                                       const float* __restrict__ fc_w,
                                                       float* __restrict__ acc) {
    const int blk  = blockIdx.x;
    const int b    = blk / (Hsz / 16);
    const int tile = blk % (Hsz / 16);
    const int lane = threadIdx.x;
    const int hf   = lane >> 4;     // which 16-lane half
    const int m    = lane & 15;     // row (segment) this lane covers in A
    const int n    = lane & 15;     // column within the H tile for B/C/D

    // Per-lane segment bounds for building the indicator A fragment.
    const int bos_m = bos[b * Ssz + m];
    const int end_m = (m < Ssz - 1) ? bos[b * Ssz + m + 1] : valid[b];

    const float* trow = toks + (size_t)b * Lsz * Hsz + tile * 16 + n;
    const float  wn   = fc_w[tile * 16 + n];

    v8f c = {};
    for (int k0 = 0; k0 < Lsz; k0 += 4) {
        v2f a, bm;
        // A fragment (indicator, built in registers — zero memory traffic)
        const int ka = k0 + 2 * hf;
        a.x = (ka     >= bos_m && ka     < end_m) ? 1.0f : 0.0f;
        a.y = (ka + 1 >= bos_m && ka + 1 < end_m) ? 1.0f : 0.0f;
        // B fragment: rows K = k0+hf and K = k0+2+hf of the token tile
        const int kb = k0 + hf;
        bm.x = trow[(size_t)kb * Hsz];
        bm.y = trow[(size_t)(kb + 2) * Hsz];
        c = __builtin_amdgcn_wmma_f32_16x16x4_f32(
                /*neg_a=*/false, a, /*neg_b=*/false, bm,
                /*c_mod=*/(short)0, c, /*reuse_a=*/false, /*reuse_b=*/false);
    }

    // logits partial: sum_n D[s][n] * fc_w[h0+n]; butterfly over the 16 N lanes
    float part[8];
    #pragma unroll
    for (int v = 0; v < 8; ++v) part[v] = c[v] * wn;
    #pragma unroll
    for (int off = 1; off < 16; off <<= 1) {
        #pragma unroll
        for (int v = 0; v < 8; ++v) part[v] += __shfl_xor(part[v], off, 32);
    }
    if (n == 0) {  // lanes 0 and 16 hold s = v and s = v + 8 respectively
        #pragma unroll
        for (int v = 0; v < 8; ++v)
            atomicAdd(&acc[b * Ssz + v + 8 * hf], part[v]);
    }
}

// ---------------------------------------------------------------------------
// Kernel 4: logits = acc / segment_count + fc_b
// ---------------------------------------------------------------------------
__global__ __launch_bounds__(512) void finalize_kernel(const float* __restrict__ acc,
                                                       const int* __restrict__ bos,
                                                       const int* __restrict__ valid,
                                                       const float* __restrict__ fc_b,
                                                       float* __restrict__ out) {
    const int idx = threadIdx.x;       // 0..511 == B*S
    const int b = idx >> 4;
    const int s = idx & 15;
    const int lo = bos[b * Ssz + s];
    const int hi = (s < Ssz - 1) ? bos[b * Ssz + s + 1] : valid[b];
    out[idx] = acc[idx] / (float)(hi - lo) + fc_b[0];
}

// ---------------------------------------------------------------------------
extern "C" void kernel_launch(void* const* d_in, const int* in_sizes, int n_in,
                              void* d_out, int out_size, void* d_ws, size_t ws_size,
                              hipStream_t stream) {
    const float* tokens = (const float*)d_in[0];   // [B, L, H] f32
    const int*   amask  = (const int*)  d_in[1];   // [B, L] i32
    const int*   bos    = (const int*)  d_in[2];   // [B, S] i32
    const float* fc_w   = (const float*)d_in[3];   // [H] f32
    const float* fc_b   = (const float*)d_in[4];   // [1] f32

    float* out_logits = (float*)d_out;                      // [B, S]
    int*   out_mask   = (int*)((float*)d_out + Bsz * Ssz);  // [B, L, L] i32

    // Workspace layout
    int*   ws_valid = (int*)d_ws;                            // B ints
    int*   ws_codes = ws_valid + Bsz;                        // B*L ints
    float* ws_acc   = (float*)(ws_codes + Bsz * Lsz);        // B*S floats

    prep_kernel<<<Bsz, 256, 0, stream>>>(amask, bos, ws_valid, ws_codes, ws_acc);

    mask_kernel<<<dim3(Lsz, Bsz), 256, 0, stream>>>(ws_codes, out_mask);

    pool_wmma_kernel<<<Bsz * (Hsz / 16), 32, 0, stream>>>(tokens, bos, ws_valid,
                                                          fc_w, ws_acc);

    finalize_kernel<<<1, 512, 0, stream>>>(ws_acc, bos, ws_valid, fc_b, out_logits);
}